// AveragePrecision_4690104287320
// MI455X (gfx1250) — compile-verified
//
#include <hip/hip_runtime.h>

// ---------------------------------------------------------------------------
// YOLO-style detection fusion for MI455X (gfx1250).
// Memory-bound: 267 MB read / 22 MB write => ~12.4 us floor at 23.3 TB/s.
// No matmul => WMMA not applicable; use CDNA5 async global->LDS staging
// (ASYNCcnt, GLOBAL_LOAD_ASYNC_TO_LDS_B128) for coalescing instead.
// ---------------------------------------------------------------------------

#define NPRED   128     // predictions per block
#define NTHREAD 128     // threads per block (4 waves of 32)
#define VDIM    85
#define NCLS    80
#define OUT7    7
#define N_TOTAL (16 * 3 * 128 * 128)            // 786432 predictions
#define PRED_OUT_ELEMS ((size_t)N_TOTAL * OUT7) // 5505024
#define TGT_ROWS (16 * 50)

typedef __attribute__((ext_vector_type(4))) float f4;
typedef __attribute__((ext_vector_type(4))) int   v4i;

// Pointer types matching the builtin's prototype:
//   void __builtin_amdgcn_global_load_async_to_lds_b128(v4i AS1*, v4i AS3*, imm, imm)
typedef __attribute__((address_space(1))) v4i* as1_v4i_ptr;
typedef __attribute__((address_space(3))) v4i* as3_v4i_ptr;

// 32-bit LDS byte offset of a generic pointer known to point into LDS.
__device__ __forceinline__ unsigned lds_off(const void* p) {
  return (unsigned)(__SIZE_TYPE__)(const __attribute__((address_space(3))) void*)p;
}

// Async 16-byte global -> LDS copy (GLOBAL_LOAD_ASYNC_TO_LDS_B128, ASYNCcnt).
__device__ __forceinline__ void async_copy16(const void* g, void* l) {
#if __has_builtin(__builtin_amdgcn_global_load_async_to_lds_b128)
  __builtin_amdgcn_global_load_async_to_lds_b128(
      (as1_v4i_ptr)g, (as3_v4i_ptr)l, /*offset=*/0, /*cpol=*/0);
#else
  asm volatile("global_load_async_to_lds_b128 %0, %1, off"
               :
               : "v"(lds_off(l)), "v"((unsigned long long)(__SIZE_TYPE__)g)
               : "memory");
#endif
}

__device__ __forceinline__ void wait_async0() {
#if __has_builtin(__builtin_amdgcn_s_wait_asynccnt)
  __builtin_amdgcn_s_wait_asynccnt(0);
#else
  asm volatile("s_wait_asynccnt 0" ::: "memory");
#endif
}

__global__ __launch_bounds__(NTHREAD)
void yolo_fuse_kernel(const float* __restrict__ outp,
                      float* __restrict__ pred_out) {
  __shared__ __align__(16) float s_in[NPRED * VDIM];   // 43,520 B
  __shared__ __align__(16) float s_out[NPRED * OUT7];  //  3,584 B

  const int tid = threadIdx.x;
  const int blk = blockIdx.x;

  // ---- Stage-in: 128 preds * 85 f32 = 2720 float4, coalesced, async to LDS.
  const f4* g4 = (const f4*)(outp + (size_t)blk * NPRED * VDIM);
  f4* s4 = (f4*)s_in;
#pragma unroll 1
  for (int i = tid; i < (NPRED * VDIM) / 4; i += NTHREAD) {
    async_copy16((const void*)(g4 + i), (void*)(s4 + i));
  }
  wait_async0();
  __syncthreads();

  // ---- Per-thread fuse from LDS (bank-conflict-free: stride 85, gcd(21,64)=1).
  const float* s = s_in + tid * VDIM;
  const float bx = s[0], by = s[1], bw = s[2], bh = s[3], obj = s[4];
  float best = s[5];
  int bid = 0;
#pragma unroll
  for (int c = 1; c < NCLS; ++c) {
    const float v = s[5 + c];
    if (v > best) { best = v; bid = c; }   // strict '>' == jnp.argmax first-max
  }
  const float conf = best * obj;
  const bool keep = conf > 0.25f;

  float* o = s_out + tid * OUT7;
  o[0] = keep ? bx : 0.0f;
  o[1] = keep ? by : 0.0f;
  o[2] = keep ? bw : 0.0f;
  o[3] = keep ? bh : 0.0f;
  o[4] = keep ? obj : 0.0f;
  o[5] = keep ? (float)bid : 0.0f;
  o[6] = keep ? conf : 0.0f;
  __syncthreads();

  // ---- Stage-out: 128*7 = 896 floats = 224 float4, coalesced nontemporal.
  const f4* so4 = (const f4*)s_out;
  f4* po4 = (f4*)(pred_out + (size_t)blk * NPRED * OUT7);
#pragma unroll 1
  for (int i = tid; i < (NPRED * OUT7) / 4; i += NTHREAD) {
    __builtin_nontemporal_store(so4[i], po4 + i);
  }
}

__global__ __launch_bounds__(256)
void targets_mask_kernel(const float* __restrict__ tgt,
                         float* __restrict__ out) {
  const int r = blockIdx.x * blockDim.x + threadIdx.x;
  if (r >= TGT_ROWS) return;
  const float* t = tgt + (size_t)r * 5;
  const float v0 = t[0], v1 = t[1], v2 = t[2], v3 = t[3], v4 = t[4];
  const bool valid =
      (v0 != 0.0f) || (v1 != 0.0f) || (v2 != 0.0f) || (v3 != 0.0f) || (v4 != 0.0f);
  float* o = out + (size_t)r * 5;
  o[0] = valid ? v0 : 0.0f;
  o[1] = valid ? v1 : 0.0f;
  o[2] = valid ? v2 : 0.0f;
  o[3] = valid ? v3 : 0.0f;
  o[4] = valid ? v4 : 0.0f;
}

extern "C" void kernel_launch(void* const* d_in, const int* in_sizes, int n_in,
                              void* d_out, int out_size, void* d_ws, size_t ws_size,
                              hipStream_t stream) {
  (void)in_sizes; (void)n_in; (void)out_size; (void)d_ws; (void)ws_size;

  const float* output  = (const float*)d_in[0];  // (16,3,128,128,85) f32
  // d_in[1] = anchors (3,2) — unused by the reference math.
  const float* targets = (const float*)d_in[2];  // (16,50,5) f32

  float* pred_out = (float*)d_out;                   // (16,49152,7) flat
  float* tgt_out  = (float*)d_out + PRED_OUT_ELEMS;  // (16,50,5) flat

  const int nblocks = N_TOTAL / NPRED;  // 6144
  yolo_fuse_kernel<<<nblocks, NTHREAD, 0, stream>>>(output, pred_out);
  targets_mask_kernel<<<(TGT_ROWS + 255) / 256, 256, 0, stream>>>(targets, tgt_out);
}